// PhysiologicalGNN_86998857548298
// MI455X (gfx1250) — compile-verified
//
#include <hip/hip_runtime.h>
#include <hip/hip_bf16.h>
#include <math.h>

// ---------------------------------------------------------------------------
// PhysiologicalGNN forward for MI455X (gfx1250, wave32, WMMA).
// All dense matrix work runs through V_WMMA_F32_16X16X4_F32 (fp32-exact path).
// Graph scatters use L2-resident f32 atomics (whole working set < 192MB L2).
// Node GEMMs compute a 16x64 tile per wave: one A-fragment feeds 4 WMMAs;
// all fragment loads are hoisted ahead of the WMMA quad (1 wait / 4 WMMAs).
// ---------------------------------------------------------------------------

#define N_NODES 50000
#define N_EDGES 800000

typedef float v2f __attribute__((ext_vector_type(2)));
typedef float v8f __attribute__((ext_vector_type(8)));

static __device__ __forceinline__ v8f v8f_zero() {
  v8f z = {0.f, 0.f, 0.f, 0.f, 0.f, 0.f, 0.f, 0.f};
  return z;
}

// ------------------------- small utility kernels ---------------------------

__global__ void gnn_fill_f32(float* __restrict__ p, float v, int n) {
  int i = blockIdx.x * blockDim.x + threadIdx.x;
  if (i < n) p[i] = v;
}

__global__ void gnn_fill_i32(int* __restrict__ p, int v, int n) {
  int i = blockIdx.x * blockDim.x + threadIdx.x;
  if (i < n) p[i] = v;
}

// d = emb[drug] @ adapt_w + adapt_b ; dproj[j] = sum_k d[k]*proj_w[5+k][j]+proj_b[j]
// vvec: per-GAT-layer per-head 3-vector  v[h][k] = sum_c lin_edge_w[k][h*64+c]*att_edge[h][c]
__global__ void gnn_prep(const float* __restrict__ emb, const int* __restrict__ drug,
                         const float* __restrict__ aw, const float* __restrict__ ab,
                         const float* __restrict__ proj_w, const float* __restrict__ proj_b,
                         const float* __restrict__ lew0, const float* __restrict__ ae0,
                         const float* __restrict__ lew1, const float* __restrict__ ae1,
                         const float* __restrict__ lew2, const float* __restrict__ ae2,
                         float* __restrict__ dproj, float* __restrict__ vvec) {
  __shared__ float d[64];
  const int t = threadIdx.x; // 64 threads
  const float* er = emb + (size_t)drug[0] * 64;
  float s = ab[t];
  for (int k = 0; k < 64; ++k) s += er[k] * aw[k * 64 + t];
  d[t] = s;
  __syncthreads();
  float p = proj_b[t];
  for (int k = 0; k < 64; ++k) p += d[k] * proj_w[(5 + k) * 64 + t];
  dproj[t] = p;
  if (t < 12) {            // layer0: H=4, HC=256
    int h = t / 3, k = t - h * 3;
    float v = 0.f;
    for (int c = 0; c < 64; ++c) v += lew0[k * 256 + h * 64 + c] * ae0[h * 64 + c];
    vvec[t] = v;
  } else if (t < 24) {     // layer1: H=4, HC=256
    int tt = t - 12, h = tt / 3, k = tt - h * 3;
    float v = 0.f;
    for (int c = 0; c < 64; ++c) v += lew1[k * 256 + h * 64 + c] * ae1[h * 64 + c];
    vvec[t] = v;
  } else if (t < 27) {     // layer2: H=1, HC=64
    int k = t - 24;
    float v = 0.f;
    for (int c = 0; c < 64; ++c) v += lew2[k * 64 + c] * ae2[c];
    vvec[t] = v;
  }
}

// h0[n,j] = sum_{k<5} x[n,k]*proj_w[k][j] + dproj[j]
__global__ void gnn_proj(const float* __restrict__ x, const float* __restrict__ proj_w,
                         const float* __restrict__ dproj, float* __restrict__ h0, int Nn) {
  int gid = blockIdx.x * blockDim.x + threadIdx.x;
  if (gid >= Nn * 64) return;
  int n = gid >> 6, j = gid & 63;
  const float* xr = x + (size_t)n * 5;
  float s = dproj[j];
  for (int k = 0; k < 5; ++k) s += xr[k] * proj_w[k * 64 + j];
  h0[gid] = s;
}

// ----------------------- generic fp32 WMMA GEMM ----------------------------
// C[M,Ncol] = act(A[M,K] @ B[K,Ncol] + bias) + res.
// One 16x64 tile per wave; per k-step: clause of 9 loads, then 4 WMMAs
// back-to-back into independent accumulators. Pointer-incremented addressing.
// Requires Ncol % 64 == 0 (true for 64/128/256 in this model).
__global__ void __launch_bounds__(256)
gnn_wmma_gemm_f32(const float* __restrict__ A, const float* __restrict__ B,
                  float* __restrict__ C, const float* __restrict__ bias,
                  const float* __restrict__ res, int M, int K, int Ncol, int act) {
  const int wave = (int)((blockIdx.x * blockDim.x + threadIdx.x) >> 5);
  const int lane = threadIdx.x & 31;
  const int tilesN = Ncol >> 6;            // 64-wide column tiles
  const int tM = wave / tilesN;
  const int tN = wave - tM * tilesN;
  if (tM * 16 >= M) return;                // wave-uniform guard (EXEC all-1 for WMMA)
  const int m0 = tM << 4, n0 = tN << 6;
  const int hl = lane & 15, sel = lane >> 4;
  const float* ap = A + (size_t)(m0 + hl) * K + (sel << 1);
  const float* bp = B + (size_t)(sel << 1) * Ncol + n0 + hl;
  v8f acc[4];
#pragma unroll
  for (int nb = 0; nb < 4; ++nb) acc[nb] = v8f_zero();
  for (int k = 0; k < K; k += 4) {
    v2f a;
    a.x = ap[0];
    a.y = ap[1];
    v2f bfr[4];
#pragma unroll
    for (int nb = 0; nb < 4; ++nb) {
      bfr[nb].x = bp[nb << 4];
      bfr[nb].y = bp[Ncol + (nb << 4)];
    }
#pragma unroll
    for (int nb = 0; nb < 4; ++nb)
      acc[nb] = __builtin_amdgcn_wmma_f32_16x16x4_f32(false, a, false, bfr[nb],
                                                      (short)0, acc[nb], false, false);
    ap += 4;
    bp += (size_t)Ncol << 2;
  }
#pragma unroll
  for (int nb = 0; nb < 4; ++nb) {
    const int bcol = n0 + (nb << 4) + hl;
    const float bb = bias ? bias[bcol] : 0.f;
#pragma unroll
    for (int r = 0; r < 8; ++r) {
      const int row = m0 + r + (sel << 3);
      float v = acc[nb][r] + bb;
      if (act == 1) v = fmaxf(v, 0.f);
      else if (act == 2) v = (v > 0.f) ? v : (expf(v) - 1.f);
      if (res) v += res[(size_t)row * Ncol + bcol];
      C[(size_t)row * Ncol + bcol] = v;
    }
  }
}

// --------------------------- GAT edge kernels ------------------------------

// a_src[n,h], a_dst[n,h] from hh [N, H*64]
__global__ void gnn_att_reduce(const float* __restrict__ hh, const float* __restrict__ att_src,
                               const float* __restrict__ att_dst, float* __restrict__ a_src,
                               float* __restrict__ a_dst, int Nn, int H) {
  int gid = blockIdx.x * blockDim.x + threadIdx.x;
  if (gid >= Nn * H) return;
  int n = gid / H, h = gid - n * H;
  const float* p = hh + (size_t)n * (H * 64) + h * 64;
  float s1 = 0.f, s2 = 0.f;
  for (int c = 0; c < 64; ++c) {
    float v = p[c];
    s1 += v * att_src[h * 64 + c];
    s2 += v * att_dst[h * 64 + c];
  }
  a_src[gid] = s1;
  a_dst[gid] = s2;
}

static __device__ __forceinline__ int f32_to_key(float f) {
  int r = __float_as_int(f);
  return (r >= 0) ? r : (r ^ 0x7fffffff);   // monotone float->int
}
static __device__ __forceinline__ float key_to_f32(int k) {
  return __int_as_float((k >= 0) ? k : (k ^ 0x7fffffff));
}

// alpha = leaky_relu(a_src[row]+a_dst[col]+a_edge, 0.2); atomic segment-max at col
__global__ void gnn_alpha(const float* __restrict__ a_src, const float* __restrict__ a_dst,
                          const float* __restrict__ eattr, const float* __restrict__ vv,
                          const int* __restrict__ ei, float* __restrict__ alpha,
                          int* __restrict__ amax, int E, int H) {
  int gid = blockIdx.x * blockDim.x + threadIdx.x;
  if (gid >= E * H) return;
  int e = gid / H, h = gid - e * H;
  int r = ei[e], c = ei[E + e];
  const float* ea = eattr + (size_t)e * 3;
  float aedge = ea[0] * vv[h * 3 + 0] + ea[1] * vv[h * 3 + 1] + ea[2] * vv[h * 3 + 2];
  float al = a_src[r * H + h] + a_dst[c * H + h] + aedge;
  al = (al > 0.f) ? al : 0.2f * al;
  alpha[gid] = al;
  atomicMax(&amax[c * H + h], f32_to_key(al));
}

// w = exp(alpha - amax[col]); denom[col] += w  (w overwrites alpha)
__global__ void gnn_softmax_w(float* __restrict__ alpha, const int* __restrict__ amax,
                              float* __restrict__ denom, const int* __restrict__ ei,
                              int E, int H) {
  int gid = blockIdx.x * blockDim.x + threadIdx.x;
  if (gid >= E * H) return;
  int e = gid / H, h = gid - e * H;
  int c = ei[E + e];
  float w = expf(alpha[gid] - key_to_f32(amax[c * H + h]));
  alpha[gid] = w;
  atomicAdd(&denom[c * H + h], w);
}

// out[col, :] += (w/denom[col]) * hh[row, :]   (one thread per edge-channel)
__global__ void gnn_gat_scatter(const float* __restrict__ hh, const float* __restrict__ w,
                                const float* __restrict__ denom, const int* __restrict__ ei,
                                float* __restrict__ out, int E, int H, int hcShift) {
  int gid = blockIdx.x * blockDim.x + threadIdx.x;
  const int HC = 1 << hcShift;
  if (gid >= E * HC) return;
  int e = gid >> hcShift;
  int cidx = gid & (HC - 1);
  int h = cidx >> 6;                       // C = 64 for every layer
  int r = ei[e], c = ei[E + e];
  float wn = w[e * H + h] / (denom[c * H + h] + 1e-16f);
  atomicAdd(&out[(size_t)c * HC + cidx], wn * hh[(size_t)r * HC + cidx]);
}

// out[i] = act(out[i] + bias[i & mask])
__global__ void gnn_bias_act(float* __restrict__ p, const float* __restrict__ bias,
                             int total, int mask, int act) {
  int i = blockIdx.x * blockDim.x + threadIdx.x;
  if (i >= total) return;
  float v = p[i] + bias[i & mask];
  if (act == 2) v = (v > 0.f) ? v : (expf(v) - 1.f);
  p[i] = v;
}

// -------------------- edge message MLP (WMMA from LDS) ---------------------
// wave handles 16 edges: ef[16,131] -> relu(@w1+b1)[16,64] -> (@w2+b2)[16,64]
// -> atomic scatter-add into sbuf[col]. K padded 131->132 (pad lanes read 0).
__global__ void __launch_bounds__(128)
gnn_msg_mlp_scatter(const float* __restrict__ h, const int* __restrict__ ei,
                    const float* __restrict__ eattr,
                    const float* __restrict__ w1, const float* __restrict__ b1,
                    const float* __restrict__ w2, const float* __restrict__ b2,
                    float* __restrict__ sbuf) {
  __shared__ float efL[4][16 * 132];
  __shared__ float z1L[4][16 * 64];
  const int wv = threadIdx.x >> 5;
  const int lane = threadIdx.x & 31;
  const int e0 = (blockIdx.x * 4 + wv) << 4;   // grid sized so e0+15 < N_EDGES

  for (int i = lane; i < 16 * 132; i += 32) {
    const int el = i / 132;
    const int k = i - el * 132;
    const int e = e0 + el;
    float v;
    if (k < 64)        v = h[(size_t)ei[e] * 64 + k];
    else if (k < 128)  v = h[(size_t)ei[N_EDGES + e] * 64 + (k - 64)];
    else if (k < 131)  v = eattr[(size_t)e * 3 + (k - 128)];
    else               v = 0.f;
    efL[wv][i] = v;
  }
  __syncthreads();

  const int hl = lane & 15, sel = lane >> 4;
  v8f acc[4];
#pragma unroll
  for (int nb = 0; nb < 4; ++nb) acc[nb] = v8f_zero();

  // GEMM1: [16,132(131)] @ w1[131,64]
  for (int kb = 0; kb < 132; kb += 4) {
    const int kk = kb + (sel << 1);
    v2f a;
    a.x = efL[wv][hl * 132 + kk];
    a.y = efL[wv][hl * 132 + kk + 1];
    v2f bfr[4];
#pragma unroll
    for (int nb = 0; nb < 4; ++nb) {
      const int col = (nb << 4) + hl;
      bfr[nb].x = (kk < 131) ? w1[kk * 64 + col] : 0.f;
      bfr[nb].y = (kk + 1 < 131) ? w1[(kk + 1) * 64 + col] : 0.f;
    }
#pragma unroll
    for (int nb = 0; nb < 4; ++nb)
      acc[nb] = __builtin_amdgcn_wmma_f32_16x16x4_f32(false, a, false, bfr[nb],
                                                      (short)0, acc[nb], false, false);
  }
  // relu(z1) -> LDS
#pragma unroll
  for (int nb = 0; nb < 4; ++nb) {
    const int col = (nb << 4) + hl;
    const float bb = b1[col];
#pragma unroll
    for (int r = 0; r < 8; ++r) {
      const int row = r + (sel << 3);
      z1L[wv][row * 64 + col] = fmaxf(acc[nb][r] + bb, 0.f);
    }
  }
  __syncthreads();

#pragma unroll
  for (int nb = 0; nb < 4; ++nb) acc[nb] = v8f_zero();

  // GEMM2: [16,64] @ w2[64,64]
  for (int kb = 0; kb < 64; kb += 4) {
    const int kk = kb + (sel << 1);
    v2f a;
    a.x = z1L[wv][hl * 64 + kk];
    a.y = z1L[wv][hl * 64 + kk + 1];
    v2f bfr[4];
#pragma unroll
    for (int nb = 0; nb < 4; ++nb) {
      const int col = (nb << 4) + hl;
      bfr[nb].x = w2[kk * 64 + col];
      bfr[nb].y = w2[(kk + 1) * 64 + col];
    }
#pragma unroll
    for (int nb = 0; nb < 4; ++nb)
      acc[nb] = __builtin_amdgcn_wmma_f32_16x16x4_f32(false, a, false, bfr[nb],
                                                      (short)0, acc[nb], false, false);
  }
  // scatter-add messages to destination nodes
#pragma unroll
  for (int nb = 0; nb < 4; ++nb) {
    const int col = (nb << 4) + hl;
    const float bb = b2[col];
#pragma unroll
    for (int r = 0; r < 8; ++r) {
      const int row = r + (sel << 3);
      const int e = e0 + row;
      const int dst = ei[N_EDGES + e];
      atomicAdd(&sbuf[(size_t)dst * 64 + col], acc[nb][r] + bb);
    }
  }
}

__global__ void gnn_degree(const int* __restrict__ ei, float* __restrict__ cnt, int E) {
  int e = blockIdx.x * blockDim.x + threadIdx.x;
  if (e < E) atomicAdd(&cnt[ei[E + e]], 1.0f);
}

// u = [h | sbuf/max(cnt,1)]
__global__ void gnn_build_u(const float* __restrict__ h, const float* __restrict__ sbuf,
                            const float* __restrict__ cnt, float* __restrict__ u, int Nn) {
  int gid = blockIdx.x * blockDim.x + threadIdx.x;
  if (gid >= Nn * 128) return;
  int n = gid >> 7, j = gid & 127;
  u[gid] = (j < 64) ? h[(size_t)n * 64 + j]
                    : sbuf[(size_t)n * 64 + (j - 64)] / fmaxf(cnt[n], 1.0f);
}

// predictor: out[n] = relu(h[n]@w1+b1) @ w2 + b2 ; one wave per node
__global__ void __launch_bounds__(256)
gnn_pred(const float* __restrict__ h, const float* __restrict__ w1,
         const float* __restrict__ b1, const float* __restrict__ w2,
         const float* __restrict__ b2, float* __restrict__ out, int Nn) {
  const int lane = threadIdx.x & 31;
  const int node = (int)((blockIdx.x * blockDim.x + threadIdx.x) >> 5);
  if (node >= Nn) return;
  const float* hr = h + (size_t)node * 64;
  float z = b1[lane];
  for (int k = 0; k < 64; ++k) z += hr[k] * w1[k * 32 + lane];
  z = fmaxf(z, 0.f) * w2[lane];
#pragma unroll
  for (int off = 16; off > 0; off >>= 1) z += __shfl_xor(z, off, 32);
  if (lane == 0) out[node] = z + b2[0];
}

// ------------------------------- launcher ----------------------------------

static inline int cdiv(int a, int b) { return (a + b - 1) / b; }

static inline void launch_gemm(const float* A, const float* B, float* C,
                               const float* bias, const float* res,
                               int M, int K, int Ncol, int act, hipStream_t s) {
  int waves = cdiv(M, 16) * (Ncol / 64);   // 16x64 tile per wave
  int blocks = cdiv(waves, 8);
  gnn_wmma_gemm_f32<<<blocks, 256, 0, s>>>(A, B, C, bias, res, M, K, Ncol, act);
}

extern "C" void kernel_launch(void* const* d_in, const int* in_sizes, int n_in,
                              void* d_out, int out_size, void* d_ws, size_t ws_size,
                              hipStream_t stream) {
  const int N = N_NODES, E = N_EDGES;

  // inputs in setup_inputs() dict insertion order, params flattened recursively
  const float* x      = (const float*)d_in[0];
  const int*   ei     = (const int*)d_in[1];      // [2,E] row-major
  const float* eattr  = (const float*)d_in[2];    // [E,3]
  const int*   drug   = (const int*)d_in[3];
  const float* emb    = (const float*)d_in[4];
  const float* adaptw = (const float*)d_in[5];
  const float* adaptb = (const float*)d_in[6];
  const float* projw  = (const float*)d_in[7];
  const float* projb  = (const float*)d_in[8];
  // gat[l]: lin_w, att_src, att_dst, lin_edge_w, att_edge, bias  at 9+6l..
  const float* g_linw[3], *g_asrc[3], *g_adst[3], *g_lew[3], *g_aedge[3], *g_bias[3];
  for (int l = 0; l < 3; ++l) {
    int b = 9 + 6 * l;
    g_linw[l]  = (const float*)d_in[b + 0];
    g_asrc[l]  = (const float*)d_in[b + 1];
    g_adst[l]  = (const float*)d_in[b + 2];
    g_lew[l]   = (const float*)d_in[b + 3];
    g_aedge[l] = (const float*)d_in[b + 4];
    g_bias[l]  = (const float*)d_in[b + 5];
  }
  const float* msg_w1 = (const float*)d_in[27];
  const float* msg_b1 = (const float*)d_in[28];
  const float* msg_w2 = (const float*)d_in[29];
  const float* msg_b2 = (const float*)d_in[30];
  const float* upd_w1 = (const float*)d_in[31];
  const float* upd_b1 = (const float*)d_in[32];
  const float* upd_w2 = (const float*)d_in[33];
  const float* upd_b2 = (const float*)d_in[34];
  const float* pred_w1 = (const float*)d_in[35];
  const float* pred_b1 = (const float*)d_in[36];
  const float* pred_w2 = (const float*)d_in[37];
  const float* pred_b2 = (const float*)d_in[38];

  float* ws = (float*)d_ws;
  // workspace layout (floats)
  size_t oScal = 0;                                // dproj[64] + vvec[27]
  size_t oA    = 256;                              // N*256  (ping: h / gat out)
  size_t oB    = oA + (size_t)N * 256;             // N*256  (pong: hh / msg bufs)
  size_t oASrc = oB + (size_t)N * 256;             // N*4
  size_t oADst = oASrc + (size_t)N * 4;            // N*4
  size_t oAmax = oADst + (size_t)N * 4;            // N*4 (int keys)
  size_t oDen  = oAmax + (size_t)N * 4;            // N*4
  size_t oAlpha = oDen + (size_t)N * 4;            // E*4

  float* dproj = ws + oScal;
  float* vvec  = ws + oScal + 64;
  float* bufA  = ws + oA;
  float* bufB  = ws + oB;
  float* aSrc  = ws + oASrc;
  float* aDst  = ws + oADst;
  int*   amax  = (int*)(ws + oAmax);
  float* den   = ws + oDen;
  float* alpha = ws + oAlpha;

  gnn_prep<<<1, 64, 0, stream>>>(emb, drug, adaptw, adaptb, projw, projb,
                                 g_lew[0], g_aedge[0], g_lew[1], g_aedge[1],
                                 g_lew[2], g_aedge[2], dproj, vvec);
  gnn_proj<<<cdiv(N * 64, 256), 256, 0, stream>>>(x, projw, dproj, bufA, N);

  // ---- three GAT layers (h in bufA, hh in bufB, out back in bufA) ----
  const int layerH[3]  = {4, 4, 1};
  const int layerK[3]  = {64, 256, 256};
  const int layerHC[3] = {256, 256, 64};
  const int layerHCs[3] = {8, 8, 6};               // log2(HC)
  const int vOff[3] = {0, 12, 24};
  for (int l = 0; l < 3; ++l) {
    const int H = layerH[l], K = layerK[l], HC = layerHC[l];
    launch_gemm(bufA, g_linw[l], bufB, nullptr, nullptr, N, K, HC, 0, stream);
    gnn_att_reduce<<<cdiv(N * H, 256), 256, 0, stream>>>(bufB, g_asrc[l], g_adst[l],
                                                         aSrc, aDst, N, H);
    gnn_fill_i32<<<cdiv(N * H, 256), 256, 0, stream>>>(amax, (int)0x80000000, N * H);
    gnn_fill_f32<<<cdiv(N * H, 256), 256, 0, stream>>>(den, 0.f, N * H);
    gnn_alpha<<<cdiv(E * H, 256), 256, 0, stream>>>(aSrc, aDst, eattr, vvec + vOff[l],
                                                    ei, alpha, amax, E, H);
    gnn_softmax_w<<<cdiv(E * H, 256), 256, 0, stream>>>(alpha, amax, den, ei, E, H);
    gnn_fill_f32<<<cdiv(N * HC, 256), 256, 0, stream>>>(bufA, 0.f, N * HC);
    gnn_gat_scatter<<<cdiv(E * HC, 256), 256, 0, stream>>>(bufB, alpha, den, ei, bufA,
                                                           E, H, layerHCs[l]);
    gnn_bias_act<<<cdiv(N * HC, 256), 256, 0, stream>>>(bufA, g_bias[l], N * HC,
                                                        HC - 1, (l < 2) ? 2 : 0);
  }

  // ---- message step: h (bufA, width 64) ----
  float* sbuf = bufB;                       // N*64
  float* ubuf = bufB + (size_t)N * 64;      // N*128
  float* tbuf = bufB + (size_t)N * 192;     // N*64
  float* cnt  = aSrc;                       // N
  gnn_fill_f32<<<cdiv(N * 64, 256), 256, 0, stream>>>(sbuf, 0.f, N * 64);
  gnn_fill_f32<<<cdiv(N, 256), 256, 0, stream>>>(cnt, 0.f, N);
  gnn_degree<<<cdiv(E, 256), 256, 0, stream>>>(ei, cnt, E);
  gnn_msg_mlp_scatter<<<E / 64, 128, 0, stream>>>(bufA, ei, eattr, msg_w1, msg_b1,
                                                  msg_w2, msg_b2, sbuf);
  gnn_build_u<<<cdiv(N * 128, 256), 256, 0, stream>>>(bufA, sbuf, cnt, ubuf, N);
  launch_gemm(ubuf, upd_w1, tbuf, upd_b1, nullptr, N, 128, 64, 1, stream);
  launch_gemm(tbuf, upd_w2, bufA, upd_b2, bufA, N, 64, 64, 0, stream); // h += mlp

  // ---- predictor ----
  gnn_pred<<<cdiv(N * 32, 256), 256, 0, stream>>>(bufA, pred_w1, pred_b1, pred_w2,
                                                  pred_b2, (float*)d_out, N);
}